// SAGELayer_24120536334768
// MI455X (gfx1250) — compile-verified
//
#include <hip/hip_runtime.h>

// ---------------------------------------------------------------------------
// SAGE layer: scatter-add aggregation -> dual fp32 WMMA GEMM (+bias, ReLU,
// fused batch-stat reduction) -> in-place BatchNorm.
// D fixed at 128 (reference). N, E derived from in_sizes.
// d_ws layout: [agg: N*128 f32][fsum: 128 f32][fsq: 128 f32]
// h is materialized directly in d_out, then normalized in place.
// ---------------------------------------------------------------------------

typedef float v2f __attribute__((ext_vector_type(2)));
typedef float v8f __attribute__((ext_vector_type(8)));

#define FEAT 128

__global__ __launch_bounds__(256) void sage_zero(float* __restrict__ p, int n) {
    int i = blockIdx.x * 256 + threadIdx.x;
    if (i < n) p[i] = 0.0f;
}

// One edge per 32 threads; each thread moves a float4 (128 feats / 32 lanes = 4).
__global__ __launch_bounds__(256) void sage_scatter(
    const float* __restrict__ x,
    const int*   __restrict__ rows,
    const int*   __restrict__ cols,
    float*       __restrict__ agg,
    int E)
{
    int gid = blockIdx.x * 256 + threadIdx.x;
    int e = gid >> 5;
    int t = gid & 31;
    if (e >= E) return;
    int r = rows[e];
    int c = cols[e];
    const float4 v = *(const float4*)(x + (size_t)c * FEAT + t * 4);
    float* dst = agg + (size_t)r * FEAT + t * 4;
    atomicAdd(dst + 0, v.x);
    atomicAdd(dst + 1, v.y);
    atomicAdd(dst + 2, v.z);
    atomicAdd(dst + 3, v.w);
}

// One wave per (16-node tile, 16-outfeat tile). Full-precision f32 WMMA.
// acc += agg_tile x W_l^T_tile ; acc += x_tile x W_r^T_tile over K=128.
// Epilogue: +bias, ReLU, store h to d_out, atomically accumulate sum/sumsq
// per output feature for the BatchNorm pass.
__global__ __launch_bounds__(256) void sage_gemm_wmma(
    const float* __restrict__ agg,
    const float* __restrict__ x,
    const float* __restrict__ Wl,
    const float* __restrict__ Wr,
    const float* __restrict__ bl,
    const float* __restrict__ br,
    float*       __restrict__ h,
    float*       __restrict__ fsum,
    float*       __restrict__ fsq,
    int n_mtiles)
{
    const int wave   = blockIdx.x * 8 + (threadIdx.x >> 5);
    const int tile_m = wave >> 3;          // node tile (N/16 of them)
    const int tile_n = wave & 7;           // output-feature tile (128/16 = 8)
    if (tile_m >= n_mtiles) return;        // wave-uniform: EXEC stays all-1s

    const int lane = threadIdx.x & 31;
    const int m16  = lane & 15;            // A: row M / B: col N this lane owns
    const int hi   = lane >> 4;            // lane half selects K-pair {0,1} vs {2,3}

    const size_t rowA = (size_t)(tile_m * 16 + m16) * FEAT; // node row
    const size_t rowW = (size_t)(tile_n * 16 + m16) * FEAT; // weight row (out feat)
    const int koff = hi * 2;

    v8f acc = {0.f, 0.f, 0.f, 0.f, 0.f, 0.f, 0.f, 0.f};

#pragma unroll 8
    for (int k = 0; k < FEAT; k += 4) {
        v2f aA = *(const v2f*)(agg + rowA + k + koff);  // A frag (agg)
        v2f bL = *(const v2f*)(Wl  + rowW + k + koff);  // B frag (W_l^T)
        acc = __builtin_amdgcn_wmma_f32_16x16x4_f32(
                  false, aA, false, bL, (short)0, acc, false, false);
        v2f aX = *(const v2f*)(x   + rowA + k + koff);  // A frag (x)
        v2f bR = *(const v2f*)(Wr  + rowW + k + koff);  // B frag (W_r^T)
        acc = __builtin_amdgcn_wmma_f32_16x16x4_f32(
                  false, aX, false, bR, (short)0, acc, false, false);
    }

    // C/D layout: VGPR v -> row (v + 8*hi), col (lane & 15) within tile.
    const int n = tile_n * 16 + m16;
    const float bias = bl[n] + br[n];
    float s = 0.f, sq = 0.f;
#pragma unroll
    for (int v = 0; v < 8; ++v) {
        float hv = acc[v] + bias;
        hv = hv > 0.f ? hv : 0.f;                      // ReLU
        const int mrow = tile_m * 16 + v + hi * 8;
        h[(size_t)mrow * FEAT + n] = hv;
        s  += hv;
        sq += hv * hv;
    }
    atomicAdd(&fsum[n], s);
    atomicAdd(&fsq[n],  sq);
}

// In-place BatchNorm over d_out using the fused batch statistics.
__global__ __launch_bounds__(256) void sage_bn(
    float*       __restrict__ h,
    const float* __restrict__ fsum,
    const float* __restrict__ fsq,
    const float* __restrict__ gamma,
    const float* __restrict__ beta,
    int total, float invN)
{
    int i = blockIdx.x * 256 + threadIdx.x;
    if (i >= total) return;
    int n = i & (FEAT - 1);
    float mean = fsum[n] * invN;
    float var  = fsq[n] * invN - mean * mean;   // biased variance (E[h^2]-E[h]^2)
    float rstd = rsqrtf(var + 1e-5f);
    h[i] = gamma[n] * (h[i] - mean) * rstd + beta[n];
}

extern "C" void kernel_launch(void* const* d_in, const int* in_sizes, int n_in,
                              void* d_out, int out_size, void* d_ws, size_t ws_size,
                              hipStream_t stream)
{
    const float* x     = (const float*)d_in[0];
    const int*   ei    = (const int*)  d_in[1];   // [2, E]: rows then cols
    const float* Wl    = (const float*)d_in[2];
    const float* bl    = (const float*)d_in[3];
    const float* Wr    = (const float*)d_in[4];
    const float* br    = (const float*)d_in[5];
    const float* gamma = (const float*)d_in[6];
    const float* beta  = (const float*)d_in[7];

    const int N = in_sizes[0] / FEAT;   // 50000
    const int E = in_sizes[1] / 2;      // 800000

    float* agg  = (float*)d_ws;               // N*128 f32
    float* fsum = agg + (size_t)N * FEAT;     // 128 f32
    float* fsq  = fsum + FEAT;                // 128 f32
    float* h    = (float*)d_out;              // N*128 f32

    // 1) zero agg + reduction cells
    int zn = N * FEAT + 2 * FEAT;
    sage_zero<<<(zn + 255) / 256, 256, 0, stream>>>(agg, zn);

    // 2) edge scatter-add (32 lanes per edge, float4 per lane)
    long long sthreads = (long long)E * 32;
    sage_scatter<<<(int)((sthreads + 255) / 256), 256, 0, stream>>>(
        x, ei, ei + E, agg, E);

    // 3) dual WMMA GEMM + bias + ReLU + batch-stat reduction
    int n_mtiles = (N + 15) / 16;             // 3125, exact (N % 16 == 0)
    int waves    = n_mtiles * 8;
    sage_gemm_wmma<<<(waves + 7) / 8, 256, 0, stream>>>(
        agg, x, Wl, Wr, bl, br, h, fsum, fsq, n_mtiles);

    // 4) in-place BatchNorm
    int total = N * FEAT;
    sage_bn<<<(total + 255) / 256, 256, 0, stream>>>(
        h, fsum, fsq, gamma, beta, total, 1.0f / (float)N);
}